// OneDeltaPerBinModel_77945066487932
// MI455X (gfx1250) — compile-verified
//
#include <hip/hip_runtime.h>
#include <hip/hip_bf16.h>

// Model dims (fixed by reference)
#define BATCH 4096
#define N0 2048
#define N1 1000
#define N2 500
#define N3 100
#define NC 100     // num clusters / experts
#define NDIM 3
#define NCLS 12

typedef __attribute__((ext_vector_type(2))) float v2f;
typedef __attribute__((ext_vector_type(8))) float v8f;

// CDNA5 async global->LDS staging (ASYNCcnt-tracked), if toolchain exposes it.
// clang-22 types the builtin's pointers as int* in AS1 (global) / AS3 (LDS).
#if defined(__gfx1250__) && \
    __has_builtin(__builtin_amdgcn_global_load_async_to_lds_b32) && \
    __has_builtin(__builtin_amdgcn_s_wait_asynccnt)
#define USE_ASYNC_B 1
typedef __attribute__((address_space(1))) int gas_int;
typedef __attribute__((address_space(3))) int las_int;
#endif

__device__ __forceinline__ v8f wmma_f32(v2f a, v2f b, v8f c) {
  return __builtin_amdgcn_wmma_f32_16x16x4_f32(
      false, a, false, b, (short)0, c, false, false);
}

// ---------------------------------------------------------------------------
// fp32 WMMA GEMM:  Y[M,N] = act(A[M,K] @ W[K,N] + bias[N])
// Block = 256 threads = 8 waves; block tile = 128(M) x 32(N).
// Wave w owns rows [m0,m0+16) and BOTH 16-col halves of the 32-col slab
// (two accumulators -> 2 WMMAs per A-fragment load).
// B slab staged through LDS in K-chunks of 32 (async global->LDS when
// available) in a K-PAIRED layout: element (k,n) lives at
//   Bs[(k>>1)*64 + n*2 + (k&1)]
// so a WMMA B fragment {B[k][n], B[k+1][n]} is one aligned ds_load_b64
// (no register repacking). No guards in the hot loop: K%32 handled by an
// unguarded multiple-of-4 tail; ragged N handled by clamped (valid-address)
// loads, since out-of-range output columns are never stored.
// Assumes K even (rows 8B-aligned) and M % 128 == 0 (true for all calls).
// ---------------------------------------------------------------------------
__global__ void k_gemm_f32(const float* __restrict__ A,
                           const float* __restrict__ W,
                           const float* __restrict__ bias,
                           float* __restrict__ Y,
                           int M, int K, int N, int relu) {
  __shared__ float Bs[32 * 32];              // K-paired layout, see above
  const int tid  = threadIdx.x;
  const int wave = tid >> 5;
  const int lane = tid & 31;
  const int l    = lane & 15;
  const int half = lane >> 4;                // A frag: 0 -> K=0,1 | 1 -> K=2,3
  const int n0   = blockIdx.x * 32;
  const int m0   = blockIdx.y * 128 + wave * 16;
  const int col0 = n0 + l;
  const int col1 = n0 + 16 + l;
  const int col0c = (col0 < N) ? col0 : N - 1;   // clamped for loads
  const int col1c = (col1 < N) ? col1 : N - 1;

  // cooperative B staging: thread covers kk = (tid>>5) + 8*j (j=0..3),
  // nn = tid&31 (fixed per thread -> one clamp). LDS dest index for
  // (kk+8j, nn) is base + 256*j  (since 8j is even).
  const int bkk  = tid >> 5;
  const int bnn  = tid & 31;
  const int bgn  = (n0 + bnn < N) ? (n0 + bnn) : N - 1;
  const int bds  = (bkk >> 1) * 64 + bnn * 2 + (bkk & 1);   // K-paired index
  const float* __restrict__ Wb = W + (size_t)bkk * N + bgn;

  const float* __restrict__ Arow = A + (size_t)(m0 + l) * K;
  v8f acc0 = {}, acc1 = {};

  const int Kmain = K & ~31;
  for (int k0 = 0; k0 < Kmain; k0 += 32) {
    __syncthreads();
    const float* __restrict__ Wc = Wb + (size_t)k0 * N;
#ifdef USE_ASYNC_B
    #pragma unroll
    for (int j = 0; j < 4; ++j)
      __builtin_amdgcn_global_load_async_to_lds_b32(
          (gas_int*)(Wc + (size_t)(8 * j) * N),
          (las_int*)&Bs[bds + 256 * j], 0, 0);
    __builtin_amdgcn_s_wait_asynccnt(0);
#else
    #pragma unroll
    for (int j = 0; j < 4; ++j)
      Bs[bds + 256 * j] = Wc[(size_t)(8 * j) * N];
#endif
    __syncthreads();
    #pragma unroll
    for (int kk = 0; kk < 32; kk += 4) {
      v2f a = *(const v2f*)(Arow + k0 + kk + 2 * half);
      // K-paired fragment loads: single aligned b64 each
      const int rb = (kk + 2 * half) * 32;   // == ((kk+2*half)>>1)*64
      v2f b0 = *(const v2f*)&Bs[rb + l * 2];
      v2f b1 = *(const v2f*)&Bs[rb + 32 + l * 2];
      acc0 = wmma_f32(a, b0, acc0);
      acc1 = wmma_f32(a, b1, acc1);
    }
  }
  // unguarded tail: K - Kmain is a multiple of 4, < 32
  for (int k = Kmain; k < K; k += 4) {
    v2f a = *(const v2f*)(Arow + k + 2 * half);
    const float* __restrict__ W0 = W + (size_t)(k + 2 * half) * N;
    const float* __restrict__ W1 = W0 + N;
    v2f b0, b1;
    b0.x = W0[col0c];  b0.y = W1[col0c];
    b1.x = W0[col1c];  b1.y = W1[col1c];
    acc0 = wmma_f32(a, b0, acc0);
    acc1 = wmma_f32(a, b1, acc1);
  }

  const int rbase = m0 + (half ? 8 : 0);     // C/D layout: vgpr r -> M=r / r+8
  if (col0 < N) {
    float bv = bias[col0];
    #pragma unroll
    for (int r = 0; r < 8; ++r) {
      float v = acc0[r] + bv;
      if (relu) v = fmaxf(v, 0.0f);
      Y[(size_t)(rbase + r) * N + col0] = v;
    }
  }
  if (col1 < N) {
    float bv = bias[col1];
    #pragma unroll
    for (int r = 0; r < 8; ++r) {
      float v = acc1[r] + bv;
      if (relu) v = fmaxf(v, 0.0f);
      Y[(size_t)(rbase + r) * N + col1] = v;
    }
  }
}

// ---------------------------------------------------------------------------
// Expert-selection pipeline helpers
// ---------------------------------------------------------------------------
__global__ void k_zero_counts(int* counts, int* fill) {
  int i = threadIdx.x;
  if (i < NC) { counts[i] = 0; fill[i] = 0; }
}

// argmax over y1 rows (first-max wins, matching jnp.argmax) + expert histogram
__global__ void k_argmax(const float* __restrict__ y1, int* __restrict__ idx,
                         int* __restrict__ counts) {
  int b = blockIdx.x * blockDim.x + threadIdx.x;
  if (b >= BATCH) return;
  const float* row = y1 + (size_t)b * NC;
  float best = row[0];
  int bi = 0;
  #pragma unroll 4
  for (int j = 1; j < NC; ++j) {
    float v = row[j];
    if (v > best) { best = v; bi = j; }
  }
  idx[b] = bi;
  atomicAdd(&counts[bi], 1);
}

__global__ void k_scan(const int* __restrict__ counts, int* __restrict__ offsets,
                       int* __restrict__ fill) {
  if (threadIdx.x == 0) {
    int s = 0;
    for (int c = 0; c < NC; ++c) {
      offsets[c] = s;
      fill[c]    = s;
      s += counts[c];
    }
  }
}

__global__ void k_scatter(const int* __restrict__ idx, int* __restrict__ fill,
                          int* __restrict__ rowlist) {
  int b = blockIdx.x * blockDim.x + threadIdx.x;
  if (b >= BATCH) return;
  int c = idx[b];
  int pos = atomicAdd(&fill[c], 1);
  rowlist[pos] = b;
}

// ---------------------------------------------------------------------------
// Selected-expert kernel: for expert c, process 16 gathered rows per block.
//   he = relu(x_rows @ We1[c] + be1[c])   via WMMA f32 (K=2048, N=100)
//   y2 = he @ We2[c] + be2[c]             (100x3, scalar epilogue from LDS)
// Block = 128 threads = 4 waves; wave w owns the 32-col slab [32w, 32w+32)
// of N3=100 (clamped loads for the ragged last slab).
// Grid = (256 row-tiles, 100 experts); empty tiles exit immediately.
// ---------------------------------------------------------------------------
__global__ void k_expert(const float* __restrict__ x,
                         const float* __restrict__ We1,
                         const float* __restrict__ be1,
                         const float* __restrict__ We2,
                         const float* __restrict__ be2,
                         const int* __restrict__ rowlist,
                         const int* __restrict__ offsets,
                         const int* __restrict__ counts,
                         float* __restrict__ y2) {
  const int c   = blockIdx.y;
  const int cnt = counts[c];
  const int t   = blockIdx.x;
  if (t * 16 >= cnt) return;

  __shared__ int   gidx[16];
  __shared__ float he_s[16][104];            // 16 rows x N3 (padded)

  const int tid = threadIdx.x;
  if (tid < 16) {
    int rl = t * 16 + tid;
    // pad ragged tiles with the expert's first row (results discarded)
    gidx[tid] = (rl < cnt) ? rowlist[offsets[c] + rl] : rowlist[offsets[c]];
  }
  __syncthreads();

  const int wave = tid >> 5;                 // 0..3, each owns 32 cols
  const int lane = tid & 31;
  const int l    = lane & 15;
  const int half = lane >> 4;
  const int n0   = wave * 32;
  const int col0 = n0 + l;
  const int col1 = n0 + 16 + l;
  const int col0c = (col0 < N3) ? col0 : N3 - 1;
  const int col1c = (col1 < N3) ? col1 : N3 - 1;

  const float* __restrict__ Wc   = We1 + (size_t)c * N0 * N3;
  const float* __restrict__ Arow = x + (size_t)gidx[l] * N0;

  v8f acc0 = {}, acc1 = {};
  #pragma unroll 4
  for (int k = 0; k < N0; k += 4) {
    v2f a = *(const v2f*)(Arow + k + 2 * half);
    const float* __restrict__ W0 = Wc + (size_t)(k + 2 * half) * N3;
    const float* __restrict__ W1 = W0 + N3;
    v2f b0, b1;
    b0.x = W0[col0c];  b0.y = W1[col0c];
    b1.x = W0[col1c];  b1.y = W1[col1c];
    acc0 = wmma_f32(a, b0, acc0);
    acc1 = wmma_f32(a, b1, acc1);
  }

  const int rrow = half ? 8 : 0;             // C/D layout
  if (col0 < N3) {
    float bv = be1[c * N3 + col0];
    #pragma unroll
    for (int r = 0; r < 8; ++r)
      he_s[rrow + r][col0] = fmaxf(acc0[r] + bv, 0.0f);
  }
  if (col1 < N3) {
    float bv = be1[c * N3 + col1];
    #pragma unroll
    for (int r = 0; r < 8; ++r)
      he_s[rrow + r][col1] = fmaxf(acc1[r] + bv, 0.0f);
  }
  __syncthreads();

  // epilogue: y2[row, d] = he[row, :] @ We2[c][:, d] + be2[c][d]
  if (tid < 16 * NDIM) {
    int r = tid / NDIM;
    int d = tid % NDIM;
    if (t * 16 + r < cnt) {
      const float* __restrict__ W2 = We2 + (size_t)c * N3 * NDIM;
      float s = be2[c * NDIM + d];
      #pragma unroll 4
      for (int k = 0; k < N3; ++k) s += he_s[r][k] * W2[k * NDIM + d];
      y2[(size_t)gidx[r] * NDIM + d] = s;
    }
  }
}

// ---------------------------------------------------------------------------
extern "C" void kernel_launch(void* const* d_in, const int* in_sizes, int n_in,
                              void* d_out, int out_size, void* d_ws, size_t ws_size,
                              hipStream_t stream) {
  const float* x    = (const float*)d_in[0];
  const float* Wcat = (const float*)d_in[1];
  const float* bcat = (const float*)d_in[2];
  const float* Wb1  = (const float*)d_in[3];
  const float* bb1  = (const float*)d_in[4];
  const float* Wb2  = (const float*)d_in[5];
  const float* bb2  = (const float*)d_in[6];
  const float* Wb3  = (const float*)d_in[7];
  const float* bb3  = (const float*)d_in[8];
  const float* We1  = (const float*)d_in[9];
  const float* be1  = (const float*)d_in[10];
  const float* We2  = (const float*)d_in[11];
  const float* be2  = (const float*)d_in[12];

  float* out = (float*)d_out;
  float* y0 = out;                                 // [4096, 12]
  float* y1 = out + (size_t)BATCH * NCLS;          // [4096, 100]
  float* y2 = out + (size_t)BATCH * (NCLS + NC);   // [4096, 3]

  // workspace layout (floats/ints), small slack between regions
  float* h1 = (float*)d_ws;                               // [4096, 1000]
  float* h2 = h1 + (size_t)BATCH * N1 + 1024;             // [4096, 500]
  int* idx     = (int*)(h2 + (size_t)BATCH * N2 + 1024);  // [4096]
  int* counts  = idx + BATCH;                             // [128]
  int* offsets = counts + 128;                            // [128]
  int* fill    = offsets + 128;                           // [128]
  int* rowlist = fill + 128;                              // [4096]

  dim3 blk(256);
  // category head: y0 = x @ Wcat + bcat
  k_gemm_f32<<<dim3((NCLS + 31) / 32, BATCH / 128), blk, 0, stream>>>(
      x, Wcat, bcat, y0, BATCH, N0, NCLS, 0);
  // bin-head MLP
  k_gemm_f32<<<dim3((N1 + 31) / 32, BATCH / 128), blk, 0, stream>>>(
      x, Wb1, bb1, h1, BATCH, N0, N1, 1);
  k_gemm_f32<<<dim3((N2 + 31) / 32, BATCH / 128), blk, 0, stream>>>(
      h1, Wb2, bb2, h2, BATCH, N1, N2, 1);
  k_gemm_f32<<<dim3((NC + 31) / 32, BATCH / 128), blk, 0, stream>>>(
      h2, Wb3, bb3, y1, BATCH, N2, NC, 0);

  // bucket rows by argmax expert
  k_zero_counts<<<1, 256, 0, stream>>>(counts, fill);
  k_argmax<<<BATCH / 256, 256, 0, stream>>>(y1, idx, counts);
  k_scan<<<1, 32, 0, stream>>>(counts, offsets, fill);
  k_scatter<<<BATCH / 256, 256, 0, stream>>>(idx, fill, rowlist);

  // only the selected expert per row (100x FLOP cut vs reference einsum)
  k_expert<<<dim3(BATCH / 16, NC), 128, 0, stream>>>(
      x, We1, be1, We2, be2, rowlist, offsets, counts, y2);
}